// BandPass_37752762532126
// MI455X (gfx1250) — compile-verified
//
#include <hip/hip_runtime.h>

typedef __attribute__((ext_vector_type(2))) float v2f;
typedef __attribute__((ext_vector_type(8))) float v8f;

#define T_LEN 8192
#define PAD   33
#define NCH   3072
#define BLK   64          // timesteps per recurrence step
#define NBI   (T_LEN/BLK) // 128 block iterations per pass

// ws float-offset layout
#define WS_D    0                     // D0..D3  (fwd T diag blocks)    4*256
#define WS_GF   1024                  // G blocks fwd                   4*256
#define WS_VF   2048                  // V blocks fwd                   4*256
#define WS_P    3072                  // P = M^64                       256
#define WS_E    3328                  // E0..E3  (bwd T blocks, J-refl) 4*256
#define WS_GB   4352                  // G blocks bwd                   4*256
#define WS_VB   5376                  // V blocks bwd                   4*256
#define WS_PADL 6400
#define WS_PADR (6400 + NCH*PAD)

__device__ __forceinline__ float dval(int d, int ii, int jj, float b0,
                                      const float* h) {
  if (d == 0) return (ii == jj) ? b0 : (jj < ii ? h[ii - 1 - jj] : 0.f);
  return h[16 * d + ii - 1 - jj];
}

// ---------------------------------------------------------------------------
// Setup: from runtime b,a build the 64-step block matrices (and their
// time-reversed variants for the backward pass).
//   z_t = M z_{t-1} + v x_t ;  y_t = b0 x_t + e0^T z_{t-1}
//   h[d] = e0^T M^d v ; G row i = e0^T M^i ; P = M^64 ; V col j = M^(63-j) v
// ---------------------------------------------------------------------------
__global__ __launch_bounds__(256) void bp_setup(const float* __restrict__ b,
                                                const float* __restrict__ a,
                                                float* __restrict__ ws) {
  __shared__ float M[16][16];
  __shared__ float Mp[16][16];
  __shared__ float Nx[16][16];
  __shared__ float w[16];
  __shared__ float wn[16];
  __shared__ float h[64];
  const int tid = threadIdx.x;
  const int r = tid >> 4, c = tid & 15;
  const float b0 = b[0];

  float m = 0.f;
  if (c == 0 && r < 10) m = -a[r + 1];
  if (r < 9 && c == (r + 1)) m += 1.f;
  M[r][c]  = m;
  Mp[r][c] = (r == c) ? 1.f : 0.f;
  if (tid < 16) w[tid] = (tid < 10) ? (b[tid + 1] - b0 * a[tid + 1]) : 0.f;
  __syncthreads();

  // w_d = M^d v : V columns (fwd col 63-d, bwd col d) and h[d]
  for (int d = 0; d < 64; ++d) {
    if (tid < 16) {
      const int mf = 63 - d;
      ws[WS_VF + (mf >> 4) * 256 + tid * 16 + (mf & 15)] = w[tid];
      ws[WS_VB + (d  >> 4) * 256 + tid * 16 + (d  & 15)] = w[tid];
      if (tid == 0) h[d] = w[0];
      float s = 0.f;
      for (int k = 0; k < 16; ++k) s += M[tid][k] * w[k];
      wn[tid] = s;
    }
    __syncthreads();
    if (tid < 16) w[tid] = wn[tid];
    __syncthreads();
  }

  // Mp = M^i : G rows (fwd straight, bwd J-reflected) and P = M^64
  for (int i = 0; i < 64; ++i) {
    if (r == 0) {
      ws[WS_GF + (i >> 4) * 256 + (i & 15) * 16 + c] = Mp[0][c];
      ws[WS_GB + (3 - (i >> 4)) * 256 + (15 - (i & 15)) * 16 + c] = Mp[0][c];
    }
    float s = 0.f;
    for (int k = 0; k < 16; ++k) s += M[r][k] * Mp[k][c];
    Nx[r][c] = s;
    __syncthreads();
    Mp[r][c] = Nx[r][c];
    __syncthreads();
  }
  ws[WS_P + tid] = Mp[r][c];

  // T diagonal-band blocks D_d and reversed E_d[i][j] = D_d[15-i][15-j]
  for (int d = 0; d < 4; ++d) {
    ws[WS_D + d * 256 + r * 16 + c] = dval(d, r, c, b0, h);
    ws[WS_E + d * 256 + r * 16 + c] = dval(d, 15 - r, 15 - c, b0, h);
  }
}

// ---------------------------------------------------------------------------
// gfx1250 async global->LDS copy (ASYNCcnt-tracked)
// ---------------------------------------------------------------------------
__device__ __forceinline__ void async_load_b128(unsigned ldsoff, const float* gp) {
  unsigned long long ga = (unsigned long long)(uintptr_t)gp;
  asm volatile("global_load_async_to_lds_b128 %0, %1, off"
               :: "v"(ldsoff), "v"(ga) : "memory");
}
__device__ __forceinline__ void wait_async() {
  asm volatile("s_wait_asynccnt 0x0" ::: "memory");
}

// ---------------------------------------------------------------------------
// WMMA helper: K=16 via 4 chained f32 16x16x4, B operands straight from LDS.
// ---------------------------------------------------------------------------
__device__ __forceinline__ v8f mma4_row(const float A8[8], const float* row,
                                        int hh, v8f cacc) {
#pragma unroll
  for (int kk = 0; kk < 4; ++kk) {
    v2f av = { A8[2 * kk], A8[2 * kk + 1] };
    v2f bv = *(const v2f*)(row + 4 * kk + 2 * hh);
    cacc = __builtin_amdgcn_wmma_f32_16x16x4_f32(
        false, av, false, bv, (short)0, cacc, false, false);
  }
  return cacc;
}

__device__ __forceinline__ void load_afrag(const float* __restrict__ mat,
                                           float A8[8], int lane) {
  const int m = lane & 15, hh = lane >> 4;
#pragma unroll
  for (int kk = 0; kk < 4; ++kk) {
    A8[2 * kk]     = mat[m * 16 + 4 * kk + 2 * hh];
    A8[2 * kk + 1] = mat[m * 16 + 4 * kk + 2 * hh + 1];
  }
}

// scalar DF2T step for pad regions
__device__ __forceinline__ float iir_step(float x, float z[10],
                                          const float bb[11], const float aa[11]) {
  float y = bb[0] * x + z[0];
#pragma unroll
  for (int i = 0; i < 9; ++i) z[i] = bb[i + 1] * x + z[i + 1] - aa[i + 1] * y;
  z[9] = bb[10] * x - aa[10] * y;
  return y;
}

// ---------------------------------------------------------------------------
// Main: one wave per 16 channels; 64-timestep block recurrence.
// Forward:  Y_p = G_p Z + sum_{q<=p} D_{p-q} X_q ; Z' = P Z + sum_q V_q X_q
// Backward: same code with J-reflected matrices, descending base, in-place.
// ---------------------------------------------------------------------------
__global__ __launch_bounds__(32) void bp_main(const float* __restrict__ sig,
                                              const float* __restrict__ b,
                                              const float* __restrict__ a,
                                              const float* __restrict__ zi,
                                              float* __restrict__ out,
                                              float* __restrict__ ws) {
  const int lane = threadIdx.x;
  const int c0 = blockIdx.x * 16;
  const int ch = lane & 15;
  const int hh = lane >> 4;

  __shared__ float Xt[16][84];   // 64-wide tile, stride 84: aligned + no bank dup
  __shared__ float Zt[16][20];

  float bb[11], aa[11];
#pragma unroll
  for (int i = 0; i < 11; ++i) { bb[i] = b[i]; aa[i] = a[i]; }
  float ziv[10];
#pragma unroll
  for (int i = 0; i < 10; ++i) ziv[i] = zi[i];

  float AD[4][8], AG[4][8], AV[4][8], AP[8];
#pragma unroll
  for (int p = 0; p < 4; ++p) {
    load_afrag(ws + WS_D  + 256 * p, AD[p], lane);
    load_afrag(ws + WS_GF + 256 * p, AG[p], lane);
    load_afrag(ws + WS_VF + 256 * p, AV[p], lane);
  }
  load_afrag(ws + WS_P, AP, lane);

  const float* xc_base  = sig + (size_t)(c0 + ch) * T_LEN;
  float*       out_base = out + (size_t)(c0 + ch) * T_LEN;
  const float* xrow = &Xt[ch][0];
  const float* zrow = &Zt[ch][0];

  // ---------------- forward prologue: left pad (scalar) ----------------
  if (lane < 16) {
    const float* x = sig + (size_t)(c0 + lane) * T_LEN;
    float* padL = ws + WS_PADL + (size_t)(c0 + lane) * PAD;
    float x0 = x[0];
    float e0 = 2.f * x0 - x[PAD];
    float z[10];
#pragma unroll
    for (int i = 0; i < 10; ++i) z[i] = ziv[i] * e0;
    for (int t = 0; t < PAD; ++t) {
      float e = 2.f * x0 - x[PAD - t];
      padL[t] = iir_step(e, z, bb, aa);
    }
#pragma unroll
    for (int i = 0; i < 10; ++i) Zt[lane][i] = z[i];
#pragma unroll
    for (int i = 10; i < 16; ++i) Zt[lane][i] = 0.f;
  }
  __syncthreads();

  // ---------------- forward core: 128 blocks of 64 ----------------
  for (int bi = 0; bi < NBI; ++bi) {
    const float* xp = xc_base + BLK * bi + 32 * hh;
    __builtin_prefetch(xp + 4096, 0, 3);
#pragma unroll
    for (int u = 0; u < 8; ++u)
      async_load_b128((unsigned)(uintptr_t)&Xt[ch][32 * hh + 4 * u], xp + 4 * u);
    wait_async();
    __syncthreads();

    v8f Zn = {};
#pragma unroll
    for (int q = 0; q < 4; ++q) Zn = mma4_row(AV[q], xrow + 16 * q, hh, Zn);
    Zn = mma4_row(AP, zrow, hh, Zn);   // only Z-dependent link in the chain

#pragma unroll
    for (int p = 0; p < 4; ++p) {
      v8f Y = {};
      Y = mma4_row(AG[p], zrow, hh, Y);
#pragma unroll
      for (int q = 0; q <= p; ++q) Y = mma4_row(AD[p - q], xrow + 16 * q, hh, Y);
      *(float4*)(out_base + BLK * bi + 16 * p + 8 * hh)     = make_float4(Y[0], Y[1], Y[2], Y[3]);
      *(float4*)(out_base + BLK * bi + 16 * p + 8 * hh + 4) = make_float4(Y[4], Y[5], Y[6], Y[7]);
    }
    __syncthreads();   // all zrow reads done before Z update
    *(float4*)&Zt[ch][8 * hh]     = make_float4(Zn[0], Zn[1], Zn[2], Zn[3]);
    *(float4*)&Zt[ch][8 * hh + 4] = make_float4(Zn[4], Zn[5], Zn[6], Zn[7]);
  }
  __syncthreads();

  // ---------------- forward epilogue: right pad (scalar) ----------------
  if (lane < 16) {
    const float* x = sig + (size_t)(c0 + lane) * T_LEN;
    float* padR = ws + WS_PADR + (size_t)(c0 + lane) * PAD;
    float z[10];
#pragma unroll
    for (int i = 0; i < 10; ++i) z[i] = Zt[lane][i];
    float xl = x[T_LEN - 1];
    for (int j = 0; j < PAD; ++j) {
      float e = 2.f * xl - x[T_LEN - 2 - j];
      padR[j] = iir_step(e, z, bb, aa);
    }
  }
  __syncthreads();

  // ---------------- backward prologue: warm state over right pad --------------
  if (lane < 16) {
    const float* padR = ws + WS_PADR + (size_t)(c0 + lane) * PAD;
    float r0 = padR[PAD - 1];
    float z[10];
#pragma unroll
    for (int i = 0; i < 10; ++i) z[i] = ziv[i] * r0;
    for (int k = 0; k < PAD; ++k) {
      float rv = padR[PAD - 1 - k];
      (void)iir_step(rv, z, bb, aa);  // outputs fall in trimmed region
    }
#pragma unroll
    for (int i = 0; i < 10; ++i) Zt[lane][i] = z[i];
#pragma unroll
    for (int i = 10; i < 16; ++i) Zt[lane][i] = 0.f;
  }

  // swap in the time-reversed matrices
#pragma unroll
  for (int p = 0; p < 4; ++p) {
    load_afrag(ws + WS_E  + 256 * p, AD[p], lane);
    load_afrag(ws + WS_GB + 256 * p, AG[p], lane);
    load_afrag(ws + WS_VB + 256 * p, AV[p], lane);
  }
  __syncthreads();

  // ---------------- backward core: 128 blocks of 64, in-place over out --------
  for (int bi = 0; bi < NBI; ++bi) {
    const int base = T_LEN - BLK - BLK * bi;   // 8128 - 64*bi, 16B-aligned
    const float* rp = out_base + base + 32 * hh;
    if (base >= 4096) __builtin_prefetch(rp - 4096, 0, 3);
#pragma unroll
    for (int u = 0; u < 8; ++u)
      async_load_b128((unsigned)(uintptr_t)&Xt[ch][32 * hh + 4 * u], rp + 4 * u);
    wait_async();
    __syncthreads();

    v8f Zn = {};
#pragma unroll
    for (int q = 0; q < 4; ++q) Zn = mma4_row(AV[q], xrow + 16 * q, hh, Zn);
    Zn = mma4_row(AP, zrow, hh, Zn);

#pragma unroll
    for (int p = 0; p < 4; ++p) {
      v8f Y = {};
      Y = mma4_row(AG[p], zrow, hh, Y);
#pragma unroll
      for (int q = p; q < 4; ++q) Y = mma4_row(AD[q - p], xrow + 16 * q, hh, Y);
      *(float4*)(out_base + base + 16 * p + 8 * hh)     = make_float4(Y[0], Y[1], Y[2], Y[3]);
      *(float4*)(out_base + base + 16 * p + 8 * hh + 4) = make_float4(Y[4], Y[5], Y[6], Y[7]);
    }
    __syncthreads();
    *(float4*)&Zt[ch][8 * hh]     = make_float4(Zn[0], Zn[1], Zn[2], Zn[3]);
    *(float4*)&Zt[ch][8 * hh + 4] = make_float4(Zn[4], Zn[5], Zn[6], Zn[7]);
  }
}

extern "C" void kernel_launch(void* const* d_in, const int* in_sizes, int n_in,
                              void* d_out, int out_size, void* d_ws, size_t ws_size,
                              hipStream_t stream) {
  (void)in_sizes; (void)n_in; (void)out_size; (void)ws_size;
  const float* sig = (const float*)d_in[0];
  const float* b   = (const float*)d_in[1];
  const float* a   = (const float*)d_in[2];
  const float* zi  = (const float*)d_in[3];
  float* out = (float*)d_out;
  float* ws  = (float*)d_ws;

  bp_setup<<<1, 256, 0, stream>>>(b, a, ws);
  bp_main<<<NCH / 16, 32, 0, stream>>>(sig, b, a, zi, out, ws);
}